// YoloLayer_67705864454461
// MI455X (gfx1250) — compile-verified
//
#include <hip/hip_runtime.h>

// ---------------------------------------------------------------------------
// YOLO decode head for MI455X (gfx1250).
// Bandwidth-bound: 188 MB in + 186 MB out = ~375 MB -> ~16 us floor @ 23.3 TB/s.
// No matrix structure -> WMMA not applicable; optimize the data path instead:
//   - one workgroup per (b, anchor, 152-position tile); 5776/152 = 38 exact
//   - gfx1250 async global->LDS b128 copies (ASYNCcnt path), 512 B/wave-instr
//   - LDS padded 4 dwords per 4-row group: keeps 16B alignment for b128 async
//     while the class-fast read stride (612 dw == 36 mod 64) stays <=2-way
//   - conf writes: fully contiguous nontemporal b128 stores (12160 f/tile)
//   - box writes: lane-stride-16B nontemporal b128 stores (fully coalesced)
// ---------------------------------------------------------------------------

#define GLOBAL_AS __attribute__((address_space(1)))
#define LDS_AS    __attribute__((address_space(3)))

typedef int   v4i __attribute__((vector_size(16)));      // 4 x i32
typedef float fx4 __attribute__((ext_vector_type(4)));   // native float4

namespace yolo {
constexpr int B  = 32;
constexpr int A  = 3;
constexpr int C  = 80;          // classes
constexpr int CH = 5 + C;       // 85 channels per anchor
constexpr int H  = 76;
constexpr int W  = 76;
constexpr int HW = H * W;       // 5776
constexpr int N  = A * HW;      // 17328 boxes per batch
constexpr int TILE    = 152;    // 2 grid rows per tile
constexpr int NTILES  = HW / TILE;       // 38 (exact, no tail)
constexpr int NCHUNK  = CH * (TILE / 4); // 3230 b128 chunks per tile
constexpr int BOXES_FLOATS = B * N * 4;  // 2,217,984
constexpr int CONF_F4_PER_TILE = TILE * C / 4;  // 3040
// LDS row base: 4-dword pad after every 4 rows -> 16B-aligned, low-conflict
constexpr int LDS_DWORDS = 84 * TILE + (84 >> 2) * 4 + TILE;  // 13004 (52,016 B)
}

__device__ __forceinline__ int rowoff(int c) {
    // dword offset of channel row c in LDS
    return c * yolo::TILE + (c >> 2) * 4;
}

__device__ __forceinline__ float fsig(float x) {
    // sigmoid via fast exp + fast reciprocal (v_exp_f32 / v_rcp_f32)
    return __frcp_rn(1.0f + __expf(-x));
}

__global__ __launch_bounds__(256)
void yolo_decode_kernel(const float* __restrict__ in,
                        float* __restrict__ boxes,
                        float* __restrict__ confs) {
    using namespace yolo;
    __shared__ float sm[LDS_DWORDS];

    const int tid   = threadIdx.x;
    const int tile  = blockIdx.x;       // 0..37
    const int a     = blockIdx.y;       // 0..2
    const int b     = blockIdx.z;       // 0..31
    const int pbase = tile * TILE;

    const float* gsrc = in + (long)((b * A + a) * CH) * HW + pbase;

    // ---- Phase 1: fill 85 x 152 tile into LDS via async copy engine ---------
#if __has_builtin(__builtin_amdgcn_global_load_async_to_lds_b128)
    for (int f = tid; f < NCHUNK; f += 256) {
        const int c  = f / (TILE / 4);          // channel row
        const int p0 = (f - c * (TILE / 4)) * 4;
        __builtin_amdgcn_global_load_async_to_lds_b128(
            (GLOBAL_AS v4i*)(gsrc + (long)c * HW + p0),
            (LDS_AS v4i*)&sm[rowoff(c) + p0],
            /*offset=*/0, /*cpol=*/0);
    }
#if __has_builtin(__builtin_amdgcn_s_wait_asynccnt)
    __builtin_amdgcn_s_wait_asynccnt(0);
#elif defined(__AMDGCN__)
    asm volatile("s_wait_asynccnt 0" ::: "memory");
#endif
#elif __has_builtin(__builtin_amdgcn_global_load_async_to_lds_b32)
    for (int e = tid; e < CH * TILE; e += 256) {
        const int c = e / TILE;
        const int p = e - c * TILE;
        __builtin_amdgcn_global_load_async_to_lds_b32(
            (GLOBAL_AS int*)(gsrc + (long)c * HW + p),
            (LDS_AS int*)&sm[rowoff(c) + p],
            /*offset=*/0, /*cpol=*/0);
    }
#if __has_builtin(__builtin_amdgcn_s_wait_asynccnt)
    __builtin_amdgcn_s_wait_asynccnt(0);
#elif defined(__AMDGCN__)
    asm volatile("s_wait_asynccnt 0" ::: "memory");
#endif
#else
    // Fallback: synchronous load + ds store (still coalesced)
    for (int e = tid; e < CH * TILE; e += 256) {
        const int c = e / TILE;
        const int p = e - c * TILE;
        sm[rowoff(c) + p] = gsrc[(long)c * HW + p];
    }
#endif
    __syncthreads();

    // ---- Phase 2a: boxes (channels 0..3, unpadded region, stride-1 reads) ---
    if (tid < TILE) {
        const int   pg  = pbase + tid;
        const int   gy  = pg / W;
        const int   gx  = pg - gy * W;
        const float inv = 1.0f / 76.0f;

        // anchors/stride: aw = {12,19,40}/8, ah = {16,36,28}/8
        const float aw = (a == 0) ? 1.5f : (a == 1) ? 2.375f : 5.0f;
        const float ah = (a == 0) ? 2.0f : (a == 1) ? 4.5f   : 3.5f;

        const float tx = sm[0 * TILE + tid];
        const float ty = sm[1 * TILE + tid];
        const float tw = sm[2 * TILE + tid];
        const float th = sm[3 * TILE + tid];

        // SCALE_XY == 1.0 -> bxy = sigmoid(o)
        const float bx = (fsig(tx) + (float)gx) * inv;
        const float by = (fsig(ty) + (float)gy) * inv;
        const float bw = __expf(tw) * (aw * inv);
        const float bh = __expf(th) * (ah * inv);

        const float bx1 = bx - 0.5f * bw;
        const float by1 = by - 0.5f * bh;

        fx4 v = {bx1, by1, bx1 + bw, by1 + bh};
        // boxes[b][n][0][0..3], n = a*HW + pg : lane stride 16B -> coalesced b128
        __builtin_nontemporal_store(
            v, (fx4*)(boxes + (size_t)(b * N + a * HW + pg) * 4));
    }

    // ---- Phase 2b: confs (channels 5..84, scaled by det = sigmoid(ch4)) -----
    // Tile's conf output region is 12160 contiguous floats -> b128 NT stores.
    fx4* cdst = (fx4*)(confs + (size_t)(b * N + a * HW + pbase) * C);
    for (int f4 = tid; f4 < CONF_F4_PER_TILE; f4 += 256) {
        const int p    = f4 / 20;            // 20 float4 chunks per position
        const int cls0 = (f4 - p * 20) * 4;
        const float det = fsig(sm[rowoff(4) + p]);    // mostly broadcast read
        fx4 v;
        v.x = fsig(sm[rowoff(5 + cls0 + 0) + p]) * det;  // lane stride 612 dw
        v.y = fsig(sm[rowoff(5 + cls0 + 1) + p]) * det;
        v.z = fsig(sm[rowoff(5 + cls0 + 2) + p]) * det;
        v.w = fsig(sm[rowoff(5 + cls0 + 3) + p]) * det;
        __builtin_nontemporal_store(v, &cdst[f4]);
    }
}

extern "C" void kernel_launch(void* const* d_in, const int* in_sizes, int n_in,
                              void* d_out, int out_size, void* d_ws, size_t ws_size,
                              hipStream_t stream) {
    using namespace yolo;
    (void)in_sizes; (void)n_in; (void)out_size; (void)d_ws; (void)ws_size;

    const float* in = (const float*)d_in[0];
    float* boxes = (float*)d_out;                  // (B, N, 1, 4) flat
    float* confs = (float*)d_out + BOXES_FLOATS;   // (B, N, C)   flat

    dim3 grid(NTILES, A, B);   // 38 x 3 x 32 = 3648 workgroups
    yolo_decode_kernel<<<grid, 256, 0, stream>>>(in, boxes, confs);
}